// NottinghamModel_21827023798955
// MI455X (gfx1250) — compile-verified
//
#include <hip/hip_runtime.h>
#include <math.h>
#include <stdint.h>

#define T_STEPS 128
#define BATCH   1024
#define DIN     66
#define HID     256
#define OUTD    65
#define OUTP    80        // OUTD padded to 5 WMMA tiles
#define RR      35
#define FOURH   1024
#define COEFF_M 0.5f

typedef float v2f __attribute__((ext_vector_type(2)));
typedef float v8f __attribute__((ext_vector_type(8)));

__device__ __forceinline__ float sigmoid_(float x) {
    return 1.0f / (1.0f + __expf(-x));
}

// --- gfx1250 async global->LDS staging (ASYNCcnt) ------------------------
// INST_OFFSET is added to BOTH the LDS and global addresses (ISA 08 §4.4),
// so one base pair + immediate offsets stages a whole per-thread stripe.
__device__ __forceinline__ void stage_async_vec(uint32_t lds, const float* g) {
    asm volatile("global_load_async_to_lds_b128 %0, %1, off\n\t"
                 "global_load_async_to_lds_b128 %0, %1, off offset:16"
                 :: "v"(lds), "v"(g) : "memory");
}
__device__ __forceinline__ void stage_async_scalar(uint32_t lds, const float* g) {
    asm volatile("global_load_async_to_lds_b32 %0, %1, off\n\t"
                 "global_load_async_to_lds_b32 %0, %1, off offset:4\n\t"
                 "global_load_async_to_lds_b32 %0, %1, off offset:8\n\t"
                 "global_load_async_to_lds_b32 %0, %1, off offset:12\n\t"
                 "global_load_async_to_lds_b32 %0, %1, off offset:16\n\t"
                 "global_load_async_to_lds_b32 %0, %1, off offset:20\n\t"
                 "global_load_async_to_lds_b32 %0, %1, off offset:24\n\t"
                 "global_load_async_to_lds_b32 %0, %1, off offset:28"
                 :: "v"(lds), "v"(g) : "memory");
}
__device__ __forceinline__ void wait_async0() {
#if __has_builtin(__builtin_amdgcn_s_wait_asynccnt)
    __builtin_amdgcn_s_wait_asynccnt(0);
#else
    asm volatile("s_wait_asynccnt 0x0" ::: "memory");
#endif
}

// 32-wide K-chunk of the 4-gate GEMM. `wp` = W + (chunkRow + 2*half)*4H + ncol;
// all loads use compile-time immediate offsets. A fragments come from LDS.
__device__ __forceinline__ void mma_chunk4(v8f* acc, const float (*As)[36],
                                           const float* __restrict__ wp,
                                           int mloc, int half)
{
    const float2* asrd = (const float2*)&As[mloc][2 * half];   // 8B aligned
    #pragma unroll
    for (int kk = 0; kk < 32; kk += 4) {
        float2 av = asrd[kk >> 1];
        v2f a; a.x = av.x; a.y = av.y;
        #pragma unroll
        for (int g = 0; g < 4; ++g) {
            v2f b;
            b.x = wp[kk * FOURH + g * HID];
            b.y = wp[kk * FOURH + FOURH + g * HID];
            acc[g] = __builtin_amdgcn_wmma_f32_16x16x4_f32(
                false, a, false, b, (short)0, acc[g], false, false);
        }
    }
}

// ---------------------------------------------------------------------------
// Fused LSTM layer step:  gates = [A1 | Hprev] @ W + bias ; pointwise update.
// Grid: (BATCH/64, HID/32), block = 256 (8 waves). Each wave: one 16x16 tile
// per gate block. Async-DMA double-buffered A staging, 1 barrier per chunk.
// ---------------------------------------------------------------------------
template<int KA, bool A1_VEC>
__global__ __launch_bounds__(256) void lstm_layer_kernel(
    const float* __restrict__ A1, int ldA1,
    const float* __restrict__ Hprev,                  // [BATCH, HID]
    const float* __restrict__ W,                      // [(KA+HID), 4H]
    const float* __restrict__ bias,                   // [4H]
    float* __restrict__ C,                            // in-place
    float* __restrict__ Hnew)
{
    __shared__ float As[2][64][36];

    const int tid  = threadIdx.x;
    const int lane = tid & 31;
    const int wave = tid >> 5;
    const int half = lane >> 4;
    const int l15  = lane & 15;
    const int mw   = wave & 3;
    const int nw   = wave >> 2;
    const int m0   = blockIdx.x * 64;
    const int n0   = blockIdx.y * 32;
    const int ncol = n0 + nw * 16 + l15;
    const int mloc = mw * 16 + l15;

    const int srow = tid >> 2;            // staging row 0..63
    const int scol = (tid & 3) * 8;       // staging col group (0/8/16/24)

    // LDS byte addresses for this thread's staging stripe (low 32 bits of
    // the generic pointer are the LDS offset on gfx1250).
    uint32_t ldsd[2];
    ldsd[0] = (uint32_t)(uintptr_t)&As[0][srow][scol];
    ldsd[1] = (uint32_t)(uintptr_t)&As[1][srow][scol];

    v8f acc[4];
    acc[0] = {}; acc[1] = {}; acc[2] = {}; acc[3] = {};

    constexpr int KFULL = (KA / 32) * 32;
    constexpr int KREM  = KA - KFULL;          // 2 for KA=66

    // ---- K remainder of the A1 slab first (plain staging, once) ----
    if (KREM > 0) {
        const float* sp = A1 + (size_t)(m0 + srow) * ldA1 + KFULL + scol;
        #pragma unroll
        for (int u = 0; u < 8; ++u)
            As[0][srow][scol + u] = (scol + u < KREM) ? sp[u] : 0.0f;
        __syncthreads();
        // W rows past KREM are real memory; A is zero there -> no guards.
        const float* wp = W + (size_t)(KFULL + 2 * half) * FOURH + ncol;
        const float2* asrd = (const float2*)&As[0][mloc][2 * half];
        constexpr int KSTEP4 = ((KREM + 3) / 4) * 4;
        #pragma unroll
        for (int kk = 0; kk < KSTEP4; kk += 4) {
            float2 av = asrd[kk >> 1];
            v2f a; a.x = av.x; a.y = av.y;
            #pragma unroll
            for (int g = 0; g < 4; ++g) {
                v2f b;
                b.x = wp[kk * FOURH + g * HID];
                b.y = wp[kk * FOURH + FOURH + g * HID];
                acc[g] = __builtin_amdgcn_wmma_f32_16x16x4_f32(
                    false, a, false, b, (short)0, acc[g], false, false);
            }
        }
        __syncthreads();
    }

    // ---- unified pipelined chunk list: [A1 full chunks][Hprev chunks] ----
    constexpr int NCH0 = KFULL / 32;
    constexpr int NCH1 = HID / 32;
    constexpr int NCH  = NCH0 + NCH1;

    const float* rowA = A1    + (size_t)(m0 + srow) * ldA1 + scol;
    const float* rowH = Hprev + (size_t)(m0 + srow) * HID  + scol;

    auto issue = [&](int c, int buf) {
        if (c < NCH0) {
            const float* src = rowA + c * 32;
            if (A1_VEC) stage_async_vec(ldsd[buf], src);
            else        stage_async_scalar(ldsd[buf], src);
        } else {
            stage_async_vec(ldsd[buf], rowH + (c - NCH0) * 32);
        }
    };

    issue(0, 0);
    int cur = 0;
    for (int c = 0; c < NCH; ++c) {
        wait_async0();          // own DMA for chunk c complete
        __syncthreads();        // everyone's DMA done; prev buffer free
        if (c + 1 < NCH) issue(c + 1, cur ^ 1);
        int wrow = (c < NCH0) ? c * 32 : KA + (c - NCH0) * 32;
        const float* wp = W + (size_t)(wrow + 2 * half) * FOURH + ncol;
        mma_chunk4(acc, As[cur], wp, mloc, half);
        cur ^= 1;
    }

    // pointwise LSTM update (gate order: i, j, f, o; forget_bias = 1.0)
    const float bi = bias[ncol];
    const float bj = bias[HID + ncol];
    const float bf = bias[2 * HID + ncol];
    const float bo = bias[3 * HID + ncol];
    #pragma unroll
    for (int v = 0; v < 8; ++v) {
        int mg = m0 + mw * 16 + v + 8 * half;
        size_t idx = (size_t)mg * HID + ncol;
        float c_old = C[idx];
        float ig = sigmoid_(acc[0][v] + bi);
        float jg = tanhf(acc[1][v] + bj);
        float fg = sigmoid_(acc[2][v] + bf + 1.0f);
        float og = sigmoid_(acc[3][v] + bo);
        float cn = c_old * fg + ig * jg;
        C[idx]    = cn;
        Hnew[idx] = tanhf(cn) * og;
    }
}

// Pack output_W [HID, OUTD] into zero-padded [HID, OUTP].
__global__ __launch_bounds__(256) void pad_wo_kernel(
    const float* __restrict__ Wo, float* __restrict__ Wop)
{
    int i = blockIdx.x * 256 + threadIdx.x;
    if (i < HID * OUTP) {
        int r = i / OUTP, c = i - r * OUTP;
        Wop[i] = (c < OUTD) ? Wo[r * OUTD + c] : 0.0f;
    }
}

// ---------------------------------------------------------------------------
// Projection + dual softmax + CE for one timestep.
// Grid: BATCH/64, block 128 (4 waves); wave = 16 rows x 5 N-tiles.
// ---------------------------------------------------------------------------
__global__ __launch_bounds__(128) void proj_softmax_kernel(
    int t,
    const float* __restrict__ H1,    // [BATCH, HID]
    const float* __restrict__ Wop,   // [HID, OUTP] zero-padded
    const float* __restrict__ bo,    // [OUTD]
    const int*   __restrict__ tgt,   // [BATCH, T, 2]
    float* __restrict__ probs,
    float* __restrict__ losses)
{
    __shared__ float Ls[64][81];

    const int tid  = threadIdx.x;
    const int lane = tid & 31;
    const int wave = tid >> 5;
    const int half = lane >> 4;
    const int l15  = lane & 15;
    const int m0   = blockIdx.x * 64;
    const int arow = m0 + wave * 16 + l15;

    v8f acc[5];
    acc[0] = {}; acc[1] = {}; acc[2] = {}; acc[3] = {}; acc[4] = {};

    const float2* hp  = (const float2*)(H1 + (size_t)arow * HID + 2 * half); // 8B aligned
    const float*  wp0 = Wop + (size_t)(2 * half) * OUTP + l15;

    #pragma unroll 4
    for (int ks = 0; ks < HID / 4; ++ks) {
        float2 av = hp[2 * ks];
        v2f a; a.x = av.x; a.y = av.y;
        const float* wp = wp0 + ks * 4 * OUTP;
        #pragma unroll
        for (int nt = 0; nt < 5; ++nt) {
            v2f b;
            b.x = wp[nt * 16];
            b.y = wp[OUTP + nt * 16];
            acc[nt] = __builtin_amdgcn_wmma_f32_16x16x4_f32(
                false, a, false, b, (short)0, acc[nt], false, false);
        }
    }

    // logits -> LDS
    #pragma unroll
    for (int nt = 0; nt < 5; ++nt) {
        int col = nt * 16 + l15;
        if (col < OUTD) {
            float bb = bo[col];
            #pragma unroll
            for (int v = 0; v < 8; ++v)
                Ls[wave * 16 + v + 8 * half][col] = acc[nt][v] + bb;
        }
    }
    __syncthreads();

    if (tid < 64) {
        const int brow  = m0 + tid;
        const float* lr = Ls[tid];
        const int rglob = t * BATCH + brow;     // time-major logits row
        const int tb    = rglob >> 7;           // r // T
        const int tt    = rglob & 127;          // r %  T

        // probs land at [tt, tb, :] of the reference's scrambled reshape
        float* pout = probs + ((size_t)tt * BATCH + tb) * OUTD;

        float mx = -1e30f;
        for (int c = 0; c < RR; ++c) mx = fmaxf(mx, lr[c]);
        float s = 0.0f;
        for (int c = 0; c < RR; ++c) s += __expf(lr[c] - mx);
        float inv = 1.0f / s;
        float lse = __logf(s) + mx;
        for (int c = 0; c < RR; ++c) pout[c] = __expf(lr[c] - mx) * inv;

        float mx2 = -1e30f;
        for (int c = RR; c < OUTD; ++c) mx2 = fmaxf(mx2, lr[c]);
        float s2 = 0.0f;
        for (int c = RR; c < OUTD; ++c) s2 += __expf(lr[c] - mx2);
        float inv2 = 1.0f / s2;
        float lse2 = __logf(s2) + mx2;
        for (int c = RR; c < OUTD; ++c) pout[c] = __expf(lr[c] - mx2) * inv2;

        // CE with batch-major target flatten (faithful mismatch)
        int tg0 = tgt[((size_t)tb * T_STEPS + tt) * 2 + 0];
        int tg1 = tgt[((size_t)tb * T_STEPS + tt) * 2 + 1];
        float mel_ce  = -(lr[tg0] - lse);
        float harm_ce = -(lr[RR + tg1] - lse2);
        losses[rglob] = COEFF_M * mel_ce + (1.0f - COEFF_M) * harm_ce;
    }
}

// Deterministic fixed-order loss reduction.
__global__ __launch_bounds__(256) void loss_reduce_kernel(
    const float* __restrict__ losses, float* __restrict__ out)
{
    __shared__ float red[256];
    float s = 0.0f;
    for (int i = threadIdx.x; i < T_STEPS * BATCH; i += 256) s += losses[i];
    red[threadIdx.x] = s;
    __syncthreads();
    for (int off = 128; off > 0; off >>= 1) {
        if (threadIdx.x < off) red[threadIdx.x] += red[threadIdx.x + off];
        __syncthreads();
    }
    if (threadIdx.x == 0)
        out[0] = red[0] / (float)T_STEPS / (float)BATCH;
}

__global__ void init_ws_kernel(float* __restrict__ ws, unsigned long long n)
{
    unsigned long long i = (unsigned long long)blockIdx.x * blockDim.x + threadIdx.x;
    if (i < n) ws[i] = 0.0f;
}

// ---------------------------------------------------------------------------
extern "C" void kernel_launch(void* const* d_in, const int* in_sizes, int n_in,
                              void* d_out, int out_size, void* d_ws, size_t ws_size,
                              hipStream_t stream) {
    const float* seq_input = (const float*)d_in[0];
    const int*   seq_tgt   = (const int*)  d_in[1];
    const float* W0 = (const float*)d_in[2];
    const float* b0 = (const float*)d_in[3];
    const float* W1 = (const float*)d_in[4];
    const float* b1 = (const float*)d_in[5];
    const float* Wo = (const float*)d_in[6];
    const float* bo = (const float*)d_in[7];
    float* out = (float*)d_out;

    const size_t BH = (size_t)BATCH * HID;
    float* ws     = (float*)d_ws;
    float* h0buf  = ws;                        // 2*BH ping-pong
    float* c0     = ws + 2 * BH;               // BH, in-place
    float* h1buf  = ws + 3 * BH;               // 2*BH ping-pong
    float* c1     = ws + 5 * BH;               // BH, in-place
    float* losses = ws + 6 * BH;               // T*BATCH
    float* Wop    = ws + 6 * BH + (size_t)T_STEPS * BATCH;  // HID*OUTP

    {
        unsigned long long zn = 6ull * BH;
        init_ws_kernel<<<(unsigned)((zn + 255) / 256), 256, 0, stream>>>(ws, zn);
        pad_wo_kernel<<<(HID * OUTP + 255) / 256, 256, 0, stream>>>(Wo, Wop);
    }

    dim3 lgrid(BATCH / 64, HID / 32);   // 16 x 8 workgroups
    for (int t = 0; t < T_STEPS; ++t) {
        const float* h0p = h0buf + (size_t)(t & 1) * BH;
        float*       h0n = h0buf + (size_t)((t + 1) & 1) * BH;
        const float* h1p = h1buf + (size_t)(t & 1) * BH;
        float*       h1n = h1buf + (size_t)((t + 1) & 1) * BH;

        lstm_layer_kernel<DIN, false><<<lgrid, 256, 0, stream>>>(
            seq_input + (size_t)t * DIN, T_STEPS * DIN,
            h0p, W0, b0, c0, h0n);
        lstm_layer_kernel<HID, true><<<lgrid, 256, 0, stream>>>(
            h0n, HID,
            h1p, W1, b1, c1, h1n);
        proj_softmax_kernel<<<BATCH / 64, 128, 0, stream>>>(
            t, h1n, Wop, bo, seq_tgt, out, losses);
    }

    loss_reduce_kernel<<<1, 256, 0, stream>>>(
        losses, out + (size_t)BATCH * T_STEPS * OUTD);
}